// MDCA_32856499815077
// MI455X (gfx1250) — compile-verified
//
#include <hip/hip_runtime.h>
#include <hip/hip_bf16.h>
#include <float.h>

#define BATCH 4096
#define NCLS  32000
#define T     512         // threads per block in main kernel (16 waves)
#define NB    512         // blocks in main kernel (8 rows each)
#define NVEC  (NCLS / 4)  // 8000 16B vectors per row

#ifndef __has_builtin
#define __has_builtin(x) 0
#endif

#if __has_builtin(__builtin_amdgcn_global_load_async_to_lds_b128)
#define USE_ASYNC_LDS 1
#else
#define USE_ASYNC_LDS 0
#endif

typedef int v4i __attribute__((vector_size(16)));

__device__ __forceinline__ void wait_async_zero() {
#if __has_builtin(__builtin_amdgcn_s_wait_asynccnt)
    __builtin_amdgcn_s_wait_asynccnt(0);
#else
    asm volatile("s_wait_asynccnt 0" ::: "memory");
#endif
}

// issue async (or fallback sync) copy of one 128KB row into LDS rowbuf
__device__ __forceinline__ void issue_row_copy(const float* __restrict__ src,
                                               float* __restrict__ rowbuf,
                                               int tid) {
#if USE_ASYNC_LDS
    const char* gbase = (const char*)src;
    char*       lbase = (char*)rowbuf;
    for (int v = tid; v < NVEC; v += T) {
        __builtin_amdgcn_global_load_async_to_lds_b128(
            (__attribute__((address_space(1))) v4i*)(gbase + (size_t)v * 16),
            (__attribute__((address_space(3))) v4i*)(lbase + v * 16),
            0, 0);
    }
#else
    const float4* s4 = (const float4*)src;
    float4*       r4 = (float4*)rowbuf;
    for (int v = tid; v < NVEC; v += T) r4[v] = s4[v];
#endif
}

// ---- block reductions (wave32, 16 waves) ----
__device__ __forceinline__ float block_max(float v, float* red) {
    #pragma unroll
    for (int off = 16; off > 0; off >>= 1)
        v = fmaxf(v, __shfl_xor(v, off, 32));
    const int lane = threadIdx.x & 31, wave = threadIdx.x >> 5;
    if (lane == 0) red[wave] = v;
    __syncthreads();
    if (wave == 0) {
        float w = (lane < (T / 32)) ? red[lane] : -FLT_MAX;
        #pragma unroll
        for (int off = 16; off > 0; off >>= 1)
            w = fmaxf(w, __shfl_xor(w, off, 32));
        if (lane == 0) red[0] = w;
    }
    __syncthreads();
    float r = red[0];
    __syncthreads();
    return r;
}

__device__ __forceinline__ float block_sum(float v, float* red) {
    #pragma unroll
    for (int off = 16; off > 0; off >>= 1)
        v += __shfl_xor(v, off, 32);
    const int lane = threadIdx.x & 31, wave = threadIdx.x >> 5;
    if (lane == 0) red[wave] = v;
    __syncthreads();
    if (wave == 0) {
        float w = (lane < (T / 32)) ? red[lane] : 0.0f;
        #pragma unroll
        for (int off = 16; off > 0; off >>= 1)
            w += __shfl_xor(w, off, 32);
        if (lane == 0) red[0] = w;
    }
    __syncthreads();
    float r = red[0];
    __syncthreads();
    return r;
}

// ---- main: single pass over 512 MB; row staged in LDS via async copy ----
__global__ void __launch_bounds__(T)
mdca_rows(const float* __restrict__ x, float* __restrict__ conf_sum) {
    extern __shared__ float smem[];
    float* rowbuf = smem;            // NCLS floats (128 KB)
    float* colacc = smem + NCLS;     // NCLS floats (128 KB)
    float* red    = smem + 2 * NCLS; // 32 floats

    const int tid = threadIdx.x;

    // kick off the first row's async copy immediately, overlapping with
    // the colacc zeroing below (blockIdx.x < BATCH always since NB < BATCH)
    issue_row_copy(x + (size_t)blockIdx.x * NCLS, rowbuf, tid);

    for (int i = tid; i < NCLS; i += T) colacc[i] = 0.0f;

    for (int row = blockIdx.x; row < BATCH; row += gridDim.x) {
        if (row != (int)blockIdx.x) {
            // previous iteration ended with __syncthreads(): rowbuf is free
            issue_row_copy(x + (size_t)row * NCLS, rowbuf, tid);
        }
        wait_async_zero();
        __syncthreads();

        // ---- row max ----
        float m = -FLT_MAX;
        for (int i = tid; i < NCLS; i += T) m = fmaxf(m, rowbuf[i]);
        m = block_max(m, red);

        // ---- exp + row sum (exp written back in place) ----
        float s = 0.0f;
        for (int i = tid; i < NCLS; i += T) {
            float e = __expf(rowbuf[i] - m);
            rowbuf[i] = e;
            s += e;
        }
        s = block_sum(s, red);
        float rz = 1.0f / s;

        // ---- accumulate per-block column sums (column owned by one thread) ----
        for (int i = tid; i < NCLS; i += T) colacc[i] += rowbuf[i] * rz;
        __syncthreads();  // all threads done with rowbuf before next copy
    }

    // ---- one atomic add per column per block ----
    for (int i = tid; i < NCLS; i += T) atomicAdd(&conf_sum[i], colacc[i]);
}

// ---- zero workspace ----
__global__ void mdca_zero(float* __restrict__ p, int n) {
    int i = blockIdx.x * blockDim.x + threadIdx.x;
    if (i < n) p[i] = 0.0f;
}

// ---- scatter-add target counts ----
__global__ void mdca_counts(const int* __restrict__ tgt, float* __restrict__ counts) {
    int b = blockIdx.x * blockDim.x + threadIdx.x;
    if (b < BATCH) {
        int c = tgt[b];
        if (c >= 0 && c < NCLS) atomicAdd(&counts[c], 1.0f);
    }
}

// ---- final scalar: (1/(B*C)) * sum_c |conf_sum[c] - counts[c]| ----
__global__ void __launch_bounds__(1024)
mdca_final(const float* __restrict__ conf_sum, const float* __restrict__ counts,
           float* __restrict__ out) {
    __shared__ float red[32];
    const int tid = threadIdx.x;
    float acc = 0.0f;
    for (int c = tid; c < NCLS; c += 1024)
        acc += fabsf(conf_sum[c] - counts[c]);
    #pragma unroll
    for (int off = 16; off > 0; off >>= 1)
        acc += __shfl_xor(acc, off, 32);
    const int lane = tid & 31, wave = tid >> 5;
    if (lane == 0) red[wave] = acc;
    __syncthreads();
    if (wave == 0) {
        float w = red[lane];  // 32 waves exactly
        #pragma unroll
        for (int off = 16; off > 0; off >>= 1)
            w += __shfl_xor(w, off, 32);
        if (lane == 0) out[0] = w / ((float)BATCH * (float)NCLS);
    }
}

extern "C" void kernel_launch(void* const* d_in, const int* in_sizes, int n_in,
                              void* d_out, int out_size, void* d_ws, size_t ws_size,
                              hipStream_t stream) {
    const float* x   = (const float*)d_in[0];
    const int*   tgt = (const int*)d_in[1];
    float*       out = (float*)d_out;

    float* conf_sum = (float*)d_ws;        // NCLS floats
    float* counts   = conf_sum + NCLS;     // NCLS floats

    mdca_zero<<<(2 * NCLS + 255) / 256, 256, 0, stream>>>(conf_sum, 2 * NCLS);
    mdca_counts<<<(BATCH + 255) / 256, 256, 0, stream>>>(tgt, counts);

    size_t shmem = (size_t)(2 * NCLS + 32) * sizeof(float);
    mdca_rows<<<NB, T, shmem, stream>>>(x, conf_sum);

    mdca_final<<<1, 1024, 0, stream>>>(conf_sum, counts, out);
}